// MLPEncoder_26869315403945
// MI455X (gfx1250) — compile-verified
//
#include <hip/hip_runtime.h>

typedef __attribute__((ext_vector_type(2))) float v2f;
typedef __attribute__((ext_vector_type(4))) float v4f;
typedef __attribute__((ext_vector_type(8))) float v8f;

// ---------------------------------------------------------------------------
// Kernel A: build the state table P[i,f] = relu(W1[:,i]+b1) . W2[f,:] + b2[f]
// M = STATE (1024), N = EMB (512), K = EMB (512).
// Block = 256 threads = 8 waves. Each block owns a 16-row strip of states
// (iBase), each wave owns a 64-wide slice of output features (nBase) and
// computes four 16x16 WMMA accumulator tiles over the full K.
// ---------------------------------------------------------------------------
__device__ __forceinline__ void store_tile(float* __restrict__ P,
                                           const v8f& acc,
                                           const float* __restrict__ b2,
                                           int iRow, int f, int EMB) {
  const float bias = b2[f];
#pragma unroll
  for (int v = 0; v < 8; ++v) {
    P[(size_t)(iRow + v) * EMB + f] = acc[v] + bias;
  }
}

__global__ __launch_bounds__(256) void mlp_state_table(
    const float* __restrict__ W1, const float* __restrict__ b1,
    const float* __restrict__ W2, const float* __restrict__ b2,
    float* __restrict__ P, int STATE, int EMB) {
  const int lane  = threadIdx.x & 31;
  const int wave  = threadIdx.x >> 5;
  const int r     = lane & 15;      // row-in-tile (A) / col-in-tile (B)
  const int khalf = lane >> 4;      // which K pair this lane holds
  const int iBase = blockIdx.x << 4;  // state rows [iBase, iBase+16)
  const int nBase = wave << 6;        // feature cols [nBase, nBase+64)

  v8f acc0 = {}, acc1 = {}, acc2 = {}, acc3 = {};

  for (int k = 0; k < EMB; k += 4) {
    const int e0 = k + (khalf << 1);  // this lane's K pair: e0, e0+1 (even)

    // A fragment: A[m,k'] = relu(W1[k', iBase+m] + b1[k'])
    // lanes 0-15 read 64B-contiguous (i varies), lanes 16-31 likewise.
    float a0 = W1[(size_t)e0 * STATE + iBase + r] + b1[e0];
    float a1 = W1[(size_t)(e0 + 1) * STATE + iBase + r] + b1[e0 + 1];
    v2f a;
    a.x = fmaxf(a0, 0.0f);
    a.y = fmaxf(a1, 0.0f);

    // B fragments: B[k',c] = W2[(nBase+16t+c)*EMB + k'] -- aligned float2
    const float* w2e = W2 + e0;
    v2f bf0 = *(const v2f*)(w2e + (size_t)(nBase +  0 + r) * EMB);
    v2f bf1 = *(const v2f*)(w2e + (size_t)(nBase + 16 + r) * EMB);
    v2f bf2 = *(const v2f*)(w2e + (size_t)(nBase + 32 + r) * EMB);
    v2f bf3 = *(const v2f*)(w2e + (size_t)(nBase + 48 + r) * EMB);

    // 8-arg form: (neg_a, A, neg_b, B, c_mod, C, reuse_a, reuse_b)
    acc0 = __builtin_amdgcn_wmma_f32_16x16x4_f32(false, a, false, bf0,
                                                 (short)0, acc0, false, false);
    acc1 = __builtin_amdgcn_wmma_f32_16x16x4_f32(false, a, false, bf1,
                                                 (short)0, acc1, false, false);
    acc2 = __builtin_amdgcn_wmma_f32_16x16x4_f32(false, a, false, bf2,
                                                 (short)0, acc2, false, false);
    acc3 = __builtin_amdgcn_wmma_f32_16x16x4_f32(false, a, false, bf3,
                                                 (short)0, acc3, false, false);
  }

  // C/D layout: lane L, VGPR v -> M = v + 8*(L>>4), N = L%16
  const int iRow = iBase + (khalf << 3);
  store_tile(P, acc0, b2, iRow, nBase +  0 + r, EMB);
  store_tile(P, acc1, b2, iRow, nBase + 16 + r, EMB);
  store_tile(P, acc2, b2, iRow, nBase + 32 + r, EMB);
  store_tile(P, acc3, b2, iRow, nBase + 48 + r, EMB);
}

// ---------------------------------------------------------------------------
// Kernel B: out[t,:] = P[max(x[t],0),:]  -- pure row gather, write-BW bound.
// One wave per token; P (2MB) stays L2-resident, output streamed with
// non-temporal b128 stores so it does not evict P.
// ---------------------------------------------------------------------------
__global__ __launch_bounds__(256) void gather_out(
    const int* __restrict__ x, const float* __restrict__ P,
    float* __restrict__ out, int ntok, int EMB) {
  const int tok = (blockIdx.x << 3) + (threadIdx.x >> 5);
  if (tok >= ntok) return;
  const int lane = threadIdx.x & 31;

  int idx = x[tok];
  idx = idx < 0 ? 0 : idx;  // clamp padding, as in reference

  const v4f* __restrict__ src = (const v4f*)(P + (size_t)idx * EMB);
  v4f* __restrict__ dst = (v4f*)(out + (size_t)tok * EMB);

  const int nvec = EMB >> 2;  // 128 x float4 per row
#pragma unroll 4
  for (int j = lane; j < nvec; j += 32) {
    __builtin_nontemporal_store(src[j], dst + j);
  }
}

// ---------------------------------------------------------------------------
extern "C" void kernel_launch(void* const* d_in, const int* in_sizes, int n_in,
                              void* d_out, int out_size, void* d_ws, size_t ws_size,
                              hipStream_t stream) {
  const int*   x  = (const int*)d_in[0];
  const float* W1 = (const float*)d_in[1];
  const float* b1 = (const float*)d_in[2];
  const float* W2 = (const float*)d_in[3];
  const float* b2 = (const float*)d_in[4];
  float* out = (float*)d_out;

  const int ntok  = in_sizes[0];        // B*S = 131072
  const int EMB   = in_sizes[2];        // 512
  const int STATE = in_sizes[1] / EMB;  // 1024

  float* P = (float*)d_ws;  // STATE*EMB*4 = 2 MB scratch

  // Stage 1: build 1024x512 table (0.54 GFLOP, f32 WMMA).
  mlp_state_table<<<dim3(STATE / 16), 256, 0, stream>>>(W1, b1, W2, b2, P,
                                                        STATE, EMB);
  // Stage 2: gather rows (write-bandwidth bound, ~11 us floor @ 23.3 TB/s).
  gather_out<<<dim3((ntok + 7) / 8), 256, 0, stream>>>(x, P, out, ntok, EMB);
}